// SSIMLoss_79233556677232
// MI455X (gfx1250) — compile-verified
//
#include <hip/hip_runtime.h>

typedef __attribute__((ext_vector_type(2))) float v2f;
typedef __attribute__((ext_vector_type(8))) float v8f;

#define NB 2
#define ND 128
#define NH 192
#define NW 192
#define NSLICE (NB*ND)                 // 256
#define HW (NH*NW)                     // 36864
#define NQ 5
#define NELEM ((size_t)NSLICE*(size_t)HW)   // 9437184 per quantity
#define LSTRIDE 196                    // padded LDS row stride: 196*4=784 (16B aligned rows), 196%64=4
#define SMEM_BYTES ((size_t)2*NH*LSTRIDE*sizeof(float))  // 301056 B < 320KB

__device__ __forceinline__ float bandv(int diff) {
  return (diff >= -5 && diff <= 5) ? 1.0f : 0.0f;
}

// Async global->LDS copy, 16B per lane. vdst = LDS byte offset, vaddr = 64-bit
// global address, no SADDR. Tracked on ASYNCcnt.
__device__ __forceinline__ void async_copy_b128(unsigned lds_off, const void* g) {
  asm volatile("global_load_async_to_lds_b128 %0, %1, off"
               :: "v"(lds_off), "v"(g) : "memory");
}
__device__ __forceinline__ void wait_asynccnt0() {
  asm volatile("s_wait_asynccnt 0x0" ::: "memory");
}

// ---------------------------------------------------------------------------
// Kernel 1: per (b,d) slice, per quantity q: build Q(h,w) in LDS, then
//   stage 1: W-pool  S = Q * Band   (banded 192x192, via WMMA f32 16x16x4)
//   stage 2: H-pool  O = Band * S   (via WMMA f32 16x16x4) -> global workspace
// ---------------------------------------------------------------------------
__global__ void ssim_pool_hw_wmma(const float* __restrict__ pred,
                                  const float* __restrict__ targ,
                                  float* __restrict__ out) {
  extern __shared__ float smem[];
  float* bufA = smem;                   // Q, 192 x 196
  float* bufB = smem + NH * LSTRIDE;    // W-pooled, 192 x 196

  const int tid   = threadIdx.x;
  const int slice = blockIdx.x;         // b*128 + d
  const int q     = blockIdx.y;         // quantity index

  const float* P = pred + (size_t)slice * HW;
  const float* T = targ + (size_t)slice * HW;
  float*       O = out + (size_t)q * NELEM + (size_t)slice * HW;

  // Phase A: quantity into padded LDS.
  if (q <= 1) {
    // Pure copy: use async tensor path (global -> LDS, no VGPR round trip).
    const float* src = (q == 0) ? P : T;
    for (int i = tid; i < HW / 4; i += 256) {     // 9216 x 16B chunks
      int h  = i / (NW / 4);
      int w4 = (i - h * (NW / 4)) * 4;
      unsigned lds_off = (unsigned)(uintptr_t)&bufA[h * LSTRIDE + w4];
      async_copy_b128(lds_off, src + (size_t)h * NW + w4);
    }
    wait_asynccnt0();
  } else {
    for (int i = tid; i < HW; i += 256) {
      int h = i / NW, w = i - h * NW;
      float v;
      if (q == 2)      { float p = P[i]; v = p * p; }
      else if (q == 3) { float t = T[i]; v = t * t; }
      else             v = P[i] * T[i];
      bufA[h * LSTRIDE + w] = v;
    }
  }
  __syncthreads();

  // Force wave id to a scalar register: tile indices and band-guard branches
  // become s_cbranch (uniform), keeping EXEC all-1s at every WMMA.
  const int wave = __builtin_amdgcn_readfirstlane(tid >> 5);
  const int lane = tid & 31;
  const int lo   = lane & 15;
  const int hi   = lane >> 4;

  // Precompute the 12 distinct band operands per stage (depend only on
  // rel = kt - nt (or mt - kt) and the K-chunk kc, never on the tile).
  v2f bB[3][4];   // stage 1 B operand: Band[kg, wg], kg-wg = (ri-1)*16+kc*4+2*hi-lo
  v2f aB[3][4];   // stage 2 A operand: Band[hg, kg], hg-kg = (1-ri)*16+lo-kc*4-2*hi
#pragma unroll
  for (int ri = 0; ri < 3; ++ri) {
#pragma unroll
    for (int kc = 0; kc < 4; ++kc) {
      int d1 = (ri - 1) * 16 + kc * 4 + 2 * hi - lo;
      bB[ri][kc].x = bandv(d1);
      bB[ri][kc].y = bandv(d1 + 1);
      int d2 = (1 - ri) * 16 + lo - kc * 4 - 2 * hi;
      aB[ri][kc].x = bandv(d2);
      aB[ri][kc].y = bandv(d2 - 1);
    }
  }

  // Stage 1: bufB = bufA x Band  (pool along W). 12x12 output tiles of 16x16.
  for (int tile = wave; tile < 144; tile += 8) {
    int mt = tile / 12, nt = tile - mt * 12;
    int w0 = nt * 16;
    v8f acc = {};
#pragma unroll
    for (int ri = 0; ri < 3; ++ri) {
      int kt = nt + ri - 1;
      if (kt < 0 || kt > 11) continue;
#pragma unroll
      for (int kc = 0; kc < 4; ++kc) {
        int k0 = kt * 16 + kc * 4;
        int arow = mt * 16 + lo;
        int acol = k0 + 2 * hi;
        v2f a;
        a.x = bufA[arow * LSTRIDE + acol];
        a.y = bufA[arow * LSTRIDE + acol + 1];
        acc = __builtin_amdgcn_wmma_f32_16x16x4_f32(
            false, a, false, bB[ri][kc], (short)0, acc, false, false);
      }
    }
    int col = w0 + lo;
#pragma unroll
    for (int r = 0; r < 8; ++r) {
      int row = mt * 16 + 8 * hi + r;   // D layout: lanes 0-15 -> M=r, 16-31 -> M=8+r
      bufB[row * LSTRIDE + col] = acc[r];
    }
  }
  __syncthreads();

  // Stage 2: O = Band x bufB  (pool along H)
  for (int tile = wave; tile < 144; tile += 8) {
    int mt = tile / 12, nt = tile - mt * 12;
    int h0 = mt * 16, w0 = nt * 16;
    v8f acc = {};
#pragma unroll
    for (int ri = 0; ri < 3; ++ri) {
      int kt = mt + ri - 1;
      if (kt < 0 || kt > 11) continue;
#pragma unroll
      for (int kc = 0; kc < 4; ++kc) {
        int k0 = kt * 16 + kc * 4;
        int kg = k0 + 2 * hi;
        v2f b;
        b.x = bufB[kg * LSTRIDE + w0 + lo];
        b.y = bufB[(kg + 1) * LSTRIDE + w0 + lo];
        acc = __builtin_amdgcn_wmma_f32_16x16x4_f32(
            false, aB[ri][kc], false, b, (short)0, acc, false, false);
      }
    }
    int col = w0 + lo;
#pragma unroll
    for (int r = 0; r < 8; ++r) {
      int row = h0 + 8 * hi + r;
      O[row * NW + col] = acc[r];
    }
  }
}

// ---------------------------------------------------------------------------
// Kernel 2: sliding-window pool along D for all 5 quantities + fused SSIM map
// + block reduction. One thread per (b, h, w) column.
// ---------------------------------------------------------------------------
__global__ void ssim_dpool_reduce(const float* __restrict__ S,
                                  float* __restrict__ acc) {
  __shared__ float red[256];
  const int gid = blockIdx.x * 256 + threadIdx.x;   // 0 .. 2*HW-1
  const int b  = gid / HW;
  const int hw = gid - b * HW;
  const float* base = S + (size_t)b * (size_t)ND * HW + hw;
  const size_t qs = NELEM;   // quantity stride
  const size_t ds = HW;      // depth stride

  float s[5];
#pragma unroll
  for (int q = 0; q < 5; ++q) {
    float t = 0.0f;
    for (int dd = 0; dd < 6; ++dd) t += base[(size_t)q * qs + (size_t)dd * ds];
    s[q] = t;   // window [-5,5] clipped at d=0
  }

  const float inv = 1.0f / 1331.0f;   // 11^3, count_include_pad
  const float kC1 = 1.0e-4f;          // 0.01^2
  const float kC2 = 9.0e-4f;          // 0.03^2
  float lsum = 0.0f;
  for (int d = 0; d < ND; ++d) {
    float mu_p = s[0] * inv, mu_t = s[1] * inv;
    float mpp = mu_p * mu_p, mtt = mu_t * mu_t, mpt = mu_p * mu_t;
    float sp  = s[2] * inv - mpp;
    float st  = s[3] * inv - mtt;
    float spt = s[4] * inv - mpt;
    float num = (2.0f * mpt + kC1) * (2.0f * spt + kC2);
    float den = (mpp + mtt + kC1) * (sp + st + kC2);
    lsum += num / den;
    int da = d + 6, dr = d - 5;
    if (da < ND) {
#pragma unroll
      for (int qq = 0; qq < 5; ++qq) s[qq] += base[(size_t)qq * qs + (size_t)da * ds];
    }
    if (dr >= 0) {
#pragma unroll
      for (int qq = 0; qq < 5; ++qq) s[qq] -= base[(size_t)qq * qs + (size_t)dr * ds];
    }
  }

  red[threadIdx.x] = lsum;
  __syncthreads();
  for (int off = 128; off > 0; off >>= 1) {
    if (threadIdx.x < off) red[threadIdx.x] += red[threadIdx.x + off];
    __syncthreads();
  }
  if (threadIdx.x == 0) atomicAdd(acc, red[0]);
}

__global__ void ssim_zero_acc(float* acc) {
  if (threadIdx.x == 0 && blockIdx.x == 0) *acc = 0.0f;
}

__global__ void ssim_finalize(const float* __restrict__ acc,
                              float* __restrict__ out) {
  if (threadIdx.x == 0 && blockIdx.x == 0)
    out[0] = 1.0f - (*acc) * (1.0f / 9437184.0f);
}

extern "C" void kernel_launch(void* const* d_in, const int* in_sizes, int n_in,
                              void* d_out, int out_size, void* d_ws, size_t ws_size,
                              hipStream_t stream) {
  const float* pred = (const float*)d_in[0];
  const float* targ = (const float*)d_in[1];
  float* pooled = (float*)d_ws;                                  // 5*N floats
  float* accp   = (float*)((char*)d_ws + (size_t)NQ * NELEM * sizeof(float));
  float* outp   = (float*)d_out;

  ssim_zero_acc<<<1, 1, 0, stream>>>(accp);
  ssim_pool_hw_wmma<<<dim3(NSLICE, NQ), 256, SMEM_BYTES, stream>>>(pred, targ, pooled);
  ssim_dpool_reduce<<<(NB * HW) / 256, 256, 0, stream>>>(pooled, accp);
  ssim_finalize<<<1, 1, 0, stream>>>(accp, outp);
}